// DeepseekMoE_63969242906700
// MI455X (gfx1250) — compile-verified
//
#include <hip/hip_runtime.h>
#include <math.h>

typedef float v2f __attribute__((ext_vector_type(2)));
typedef float v8f __attribute__((ext_vector_type(8)));

#define NTOK 32768
#define HDIM 128
#define NEXP 64
#define TOPK 6
#define ISZ  80
#define SHI  160
#define TM   64                       // tokens per expert tile
#define STM  32                       // tokens per shared-expert tile
#define CAP  (NTOK*TOPK + NEXP*TM)    // padded gather capacity
#define MAXTILES (CAP/TM)
#define XS   132                      // LDS stride for X rows (528 B, 16B aligned)
#define MS   81                       // LDS stride for mid (80) rows, 81%64=17 coprime
#define MSH  161                      // LDS stride for shared mid (160) rows

__device__ __forceinline__ v8f wmma4(v2f a, v2f b, v8f c) {
  // V_WMMA_F32_16X16X4_F32 : D = A(16x4) * B(4x16) + C(16x16), fp32
  return __builtin_amdgcn_wmma_f32_16x16x4_f32(false, a, false, b, (short)0, c,
                                               false, false);
}

// Async 16-byte global->LDS copy (CDNA5 TDM-adjacent path, ASYNCcnt-tracked).
__device__ __forceinline__ void async_copy_b128(const float* gsrc, float* ldst) {
  unsigned lds_off = (unsigned)(size_t)ldst;               // low 32b = LDS byte offset
  unsigned long long gaddr = (unsigned long long)(size_t)gsrc;
  asm volatile("global_load_async_to_lds_b128 %0, %1, off"
               :: "v"(lds_off), "v"(gaddr) : "memory");
}
__device__ __forceinline__ void async_wait0() {
  asm volatile("s_wait_asynccnt 0x0" ::: "memory");
}

// ---------------------------------------------------------------- init lists
__global__ void moe_init_kernel(int* __restrict__ counts,
                                int* __restrict__ gtok, float* __restrict__ gw) {
  int i = blockIdx.x * 256 + threadIdx.x;
  if (i < NEXP) counts[i] = 0;
  if (i < CAP) { gtok[i] = 0; gw[i] = 0.f; }
}

// ---------------------------------------------------------------- router
__global__ __launch_bounds__(256) void moe_router_kernel(
    const float* __restrict__ r, const float* __restrict__ rw,
    float* __restrict__ scores_out, int* __restrict__ counts,
    int* __restrict__ tpi, float* __restrict__ tpw) {
  __shared__ float lw[NEXP * HDIM];          // 32 KB of router weights
  for (int i = threadIdx.x; i < NEXP * HDIM; i += 256) lw[i] = rw[i];
  __syncthreads();

  int n = blockIdx.x * 256 + threadIdx.x;    // one token per thread
  const float* rr = r + (size_t)n * HDIM;

  float acc[NEXP];
#pragma unroll
  for (int e = 0; e < NEXP; e++) acc[e] = 0.f;
  for (int k = 0; k < HDIM; k += 4) {
    float4 rv = *(const float4*)(rr + k);
#pragma unroll
    for (int e = 0; e < NEXP; e++) {
      acc[e] = fmaf(rv.x, lw[e * HDIM + k + 0], acc[e]);
      acc[e] = fmaf(rv.y, lw[e * HDIM + k + 1], acc[e]);
      acc[e] = fmaf(rv.z, lw[e * HDIM + k + 2], acc[e]);
      acc[e] = fmaf(rv.w, lw[e * HDIM + k + 3], acc[e]);
    }
  }
  // softmax
  float mx = -1e30f;
#pragma unroll
  for (int e = 0; e < NEXP; e++) mx = fmaxf(mx, acc[e]);
  float sum = 0.f;
#pragma unroll
  for (int e = 0; e < NEXP; e++) { acc[e] = __expf(acc[e] - mx); sum += acc[e]; }
  float inv = 1.f / sum;
#pragma unroll
  for (int e = 0; e < NEXP; e++) {
    acc[e] *= inv;
    scores_out[(size_t)n * NEXP + e] = acc[e];
  }
  // top-6 by repeated unrolled argmax
  unsigned long long chosen = 0ull;
  int bi[TOPK]; float bw[TOPK]; float wsum = 0.f;
#pragma unroll
  for (int t = 0; t < TOPK; t++) {
    float best = -1.f; int bidx = 0;
#pragma unroll
    for (int e = 0; e < NEXP; e++) {
      bool ok = (((chosen >> e) & 1ull) == 0ull) && (acc[e] > best);
      best = ok ? acc[e] : best;
      bidx = ok ? e : bidx;
    }
    chosen |= (1ull << bidx);
    bi[t] = bidx; bw[t] = best; wsum += best;
  }
  float wn = 1.f / (wsum + 1e-20f);
#pragma unroll
  for (int t = 0; t < TOPK; t++) {
    tpi[n * TOPK + t] = bi[t];
    tpw[n * TOPK + t] = bw[t] * wn;
    atomicAdd(&counts[bi[t]], 1);
  }
}

// ---------------------------------------------------------------- scan
__global__ void moe_scan_kernel(const int* __restrict__ counts,
                                int* __restrict__ offs, int* __restrict__ fill) {
  if (threadIdx.x == 0) {
    int run = 0;
    for (int e = 0; e < NEXP; e++) {
      offs[e] = run;
      int p = (counts[e] + TM - 1) / TM * TM;   // pad each expert to TM tokens
      run += p;
    }
    offs[NEXP] = run;
  }
  __syncthreads();
  if (threadIdx.x < NEXP) fill[threadIdx.x] = offs[threadIdx.x];
}

// ---------------------------------------------------------------- scatter
__global__ void moe_scatter_kernel(const int* __restrict__ tpi,
                                   const float* __restrict__ tpw,
                                   int* __restrict__ fill,
                                   int* __restrict__ gtok, float* __restrict__ gw) {
  int j = blockIdx.x * 256 + threadIdx.x;
  if (j >= NTOK * TOPK) return;
  int e = tpi[j];
  int pos = atomicAdd(&fill[e], 1);
  gtok[pos] = j / TOPK;
  gw[pos] = tpw[j];
}

// ---------------------------------------------------------------- shared expert + residual (writes y)
__global__ __launch_bounds__(256) void moe_shared_kernel(
    const float* __restrict__ x, const float* __restrict__ Sg,
    const float* __restrict__ Su, const float* __restrict__ Sd,
    float* __restrict__ y) {
  __shared__ float Xs[STM * XS];
  __shared__ float Mid[STM * MSH];

  int base = blockIdx.x * STM;
  // async gather of the 32-token X tile straight into LDS
  for (int i = threadIdx.x; i < STM * HDIM / 4; i += 256) {
    int row = i >> 5; int c4 = (i & 31) << 2;
    async_copy_b128(x + (size_t)(base + row) * HDIM + c4, &Xs[row * XS + c4]);
  }
  async_wait0();
  __syncthreads();

  int lane = threadIdx.x & 31;
  int wv = threadIdx.x >> 5;
  int half = lane >> 4;
  int lr = lane & 15;

  // Stage A: Mid[32 x 160] = silu(X Sg^T) * (X Su^T)
  for (int t = wv; t < 2 * 10; t += 8) {
    int mi = t / 10, ni = t % 10;
    v8f ag = {}, au = {};
    const float* xrow = &Xs[(mi * 16 + lr) * XS + 2 * half];
    const float* gr = Sg + (size_t)(ni * 16 + lr) * HDIM + 2 * half;
    const float* ur = Su + (size_t)(ni * 16 + lr) * HDIM + 2 * half;
    for (int k = 0; k < HDIM; k += 4) {
      v2f a; a.x = xrow[k]; a.y = xrow[k + 1];
      v2f bg; bg.x = gr[k]; bg.y = gr[k + 1];
      v2f bu; bu.x = ur[k]; bu.y = ur[k + 1];
      ag = wmma4(a, bg, ag);
      au = wmma4(a, bu, au);
    }
#pragma unroll
    for (int vv = 0; vv < 8; vv++) {
      float g = ag[vv];
      float sil = g / (1.f + __expf(-g));
      int m = mi * 16 + vv + 8 * half;
      Mid[m * MSH + ni * 16 + lr] = sil * au[vv];
    }
  }
  __syncthreads();

  // Stage B: y = Mid @ Sd^T + x   (K = 160)
  for (int t = wv; t < 2 * 8; t += 8) {
    int mi = t / 8, ni = t % 8;
    v8f acc = {};
    const float* mrow = &Mid[(mi * 16 + lr) * MSH + 2 * half];
    const float* dr = Sd + (size_t)(ni * 16 + lr) * SHI + 2 * half;
    for (int k = 0; k < SHI; k += 4) {
      v2f a; a.x = mrow[k]; a.y = mrow[k + 1];
      v2f b; b.x = dr[k]; b.y = dr[k + 1];
      acc = wmma4(a, b, acc);
    }
#pragma unroll
    for (int vv = 0; vv < 8; vv++) {
      int m = mi * 16 + vv + 8 * half;
      int col = ni * 16 + lr;
      y[(size_t)(base + m) * HDIM + col] = acc[vv] + Xs[m * XS + col];
    }
  }
}

// ---------------------------------------------------------------- routed experts (atomic accumulate into y)
__global__ __launch_bounds__(256) void moe_expert_kernel(
    const float* __restrict__ x,
    const float* __restrict__ Wg, const float* __restrict__ Wu,
    const float* __restrict__ Wd,
    const int* __restrict__ offs, const int* __restrict__ gtok,
    const float* __restrict__ gw, float* __restrict__ y) {
  __shared__ float Xs[TM * XS];
  __shared__ float Mid[TM * MS];
  __shared__ int toks[TM];
  __shared__ float twt[TM];

  int base = blockIdx.x * TM;
  if (base >= offs[NEXP]) return;     // uniform: tile beyond total padded work
  int e = 0;
  for (int i = 1; i < NEXP; i++) if (offs[i] <= base) e = i;

  const float* wg = Wg + (size_t)e * ISZ * HDIM;
  const float* wu = Wu + (size_t)e * ISZ * HDIM;
  const float* wd = Wd + (size_t)e * HDIM * ISZ;

  for (int i = threadIdx.x; i < TM; i += 256) {
    toks[i] = gtok[base + i];
    twt[i] = gw[base + i];
  }
  __syncthreads();
  // async gather of the 64-token X tile straight into LDS
  for (int i = threadIdx.x; i < TM * HDIM / 4; i += 256) {
    int row = i >> 5; int c4 = (i & 31) << 2;
    async_copy_b128(x + (size_t)toks[row] * HDIM + c4, &Xs[row * XS + c4]);
  }
  async_wait0();
  __syncthreads();

  int lane = threadIdx.x & 31;
  int wv = threadIdx.x >> 5;
  int half = lane >> 4;
  int lr = lane & 15;

  // Stage A: Mid[64 x 80] = silu(X Wg^T) * (X Wu^T)
  for (int t = wv; t < 4 * 5; t += 8) {
    int mi = t / 5, ni = t % 5;
    v8f ag = {}, au = {};
    const float* xrow = &Xs[(mi * 16 + lr) * XS + 2 * half];
    const float* gr = wg + (size_t)(ni * 16 + lr) * HDIM + 2 * half;
    const float* ur = wu + (size_t)(ni * 16 + lr) * HDIM + 2 * half;
    for (int k = 0; k < HDIM; k += 4) {
      v2f a; a.x = xrow[k]; a.y = xrow[k + 1];
      v2f bg; bg.x = gr[k]; bg.y = gr[k + 1];
      v2f bu; bu.x = ur[k]; bu.y = ur[k + 1];
      ag = wmma4(a, bg, ag);
      au = wmma4(a, bu, au);
    }
#pragma unroll
    for (int vv = 0; vv < 8; vv++) {
      float g = ag[vv];
      float sil = g / (1.f + __expf(-g));
      int m = mi * 16 + vv + 8 * half;
      Mid[m * MS + ni * 16 + lr] = sil * au[vv];
    }
  }
  __syncthreads();

  // Stage B: y[tok] += w * (Mid @ Wd^T)   (K = 80)
  for (int t = wv; t < 4 * 8; t += 8) {
    int mi = t / 8, ni = t % 8;
    v8f acc = {};
    const float* mrow = &Mid[(mi * 16 + lr) * MS + 2 * half];
    const float* dr = wd + (size_t)(ni * 16 + lr) * ISZ + 2 * half;
    for (int k = 0; k < ISZ; k += 4) {
      v2f a; a.x = mrow[k]; a.y = mrow[k + 1];
      v2f b; b.x = dr[k]; b.y = dr[k + 1];
      acc = wmma4(a, b, acc);
    }
#pragma unroll
    for (int vv = 0; vv < 8; vv++) {
      int m = mi * 16 + vv + 8 * half;
      float val = acc[vv] * twt[m];
      atomicAdd(&y[(size_t)toks[m] * HDIM + ni * 16 + lr], val);
    }
  }
}

// ---------------------------------------------------------------- launcher
extern "C" void kernel_launch(void* const* d_in, const int* in_sizes, int n_in,
                              void* d_out, int out_size, void* d_ws, size_t ws_size,
                              hipStream_t stream) {
  (void)in_sizes; (void)n_in; (void)out_size; (void)ws_size;
  const float* x  = (const float*)d_in[0];   // hidden_states [N,H]
  const float* r  = (const float*)d_in[1];   // tgt_route     [N,H]
  const float* rw = (const float*)d_in[2];   // router_w      [E,H]
  const float* Wg = (const float*)d_in[3];   // [E,I,H]
  const float* Wu = (const float*)d_in[4];   // [E,I,H]
  const float* Wd = (const float*)d_in[5];   // [E,H,I]
  const float* Sg = (const float*)d_in[6];   // [SHI,H]
  const float* Su = (const float*)d_in[7];   // [SHI,H]
  const float* Sd = (const float*)d_in[8];   // [H,SHI]

  float* y      = (float*)d_out;                       // [N,H]
  float* scores = y + (size_t)NTOK * HDIM;             // [N,E]

  int*   wsi = (int*)d_ws;
  float* wsf = (float*)d_ws;
  int*   counts = wsi;                                  // 64
  int*   offs   = wsi + 64;                             // 65
  int*   fill   = wsi + 192;                            // 64
  int*   tpi    = wsi + 256;                            // N*K
  float* tpw    = wsf + 256 + NTOK * TOPK;              // N*K
  int*   gtok   = wsi + 256 + 2 * NTOK * TOPK;          // CAP
  float* gw     = wsf + 256 + 2 * NTOK * TOPK + CAP;    // CAP

  moe_init_kernel<<<(CAP + 255) / 256, 256, 0, stream>>>(counts, gtok, gw);
  moe_router_kernel<<<NTOK / 256, 256, 0, stream>>>(r, rw, scores, counts, tpi, tpw);
  moe_scan_kernel<<<1, 128, 0, stream>>>(counts, offs, fill);
  moe_scatter_kernel<<<(NTOK * TOPK + 255) / 256, 256, 0, stream>>>(tpi, tpw, fill, gtok, gw);
  moe_shared_kernel<<<NTOK / STM, 256, 0, stream>>>(x, Sg, Su, Sd, y);
  moe_expert_kernel<<<MAXTILES, 256, 0, stream>>>(x, Wg, Wu, Wd, offs, gtok, gw, y);
}